// MultiheadBlockAttention_4715874091550
// MI455X (gfx1250) — compile-verified
//
#include <hip/hip_runtime.h>
#include <hip/hip_bf16.h>

typedef __attribute__((ext_vector_type(2))) float v2f;
typedef __attribute__((ext_vector_type(8))) float v8f;

#define WMMA_F32(a, b, c) \
    __builtin_amdgcn_wmma_f32_16x16x4_f32(false, (a), false, (b), (short)0, (c), false, false)

static __device__ __forceinline__ float neg_inf() { return -__builtin_inff(); }

// ---------------------------------------------------------------------------
// Generic 256-wide GEMM:  Y[row, n] = sum_k X[row,k] * W[n,k] + bias[n]
// X is [nrows, 256] row-major (IN_HEADS=0) or head-split [B][H=4][rpb][HD=64]
// (IN_HEADS=1, row -> (b=row/rpb, r=row%rpb), k -> (h=k>>6, hd=k&63)).
// Same choice for the output layout via OUT_HEADS.
// Block: 256 threads = 8 wave32; block tile 32 rows x 256 cols;
// each wave owns a 32x32 tile = 4 WMMA accumulators (4 loads : 4 WMMAs / k-step).
// ---------------------------------------------------------------------------
template <int IN_HEADS, int OUT_HEADS>
__global__ void gemm256(const float* __restrict__ X, const float* __restrict__ W,
                        const float* __restrict__ bias, float* __restrict__ Y,
                        int nrows, int rpb)
{
    const int wave = threadIdx.x >> 5;
    const int lane = threadIdx.x & 31;
    const int row0 = blockIdx.x * 32;
    const int n0   = wave * 32;

    const int r    = lane & 15;
    const int ksel = (lane >> 4) * 2;
    const int colc = lane & 15;

    const int rA0 = row0 + r;                 // nrows is a multiple of 32 here
    const int rA1 = row0 + 16 + r;
    int Ab0 = 0, Ar0 = 0, Ab1 = 0, Ar1 = 0;
    if (IN_HEADS) {
        Ab0 = rA0 / rpb; Ar0 = rA0 % rpb;
        Ab1 = rA1 / rpb; Ar1 = rA1 % rpb;
    }

    v8f acc00 = {}, acc01 = {}, acc10 = {}, acc11 = {};

    for (int k = 0; k < 256; k += 4) {
        const int kk = k + ksel;
        v2f a0, a1;
        if (IN_HEADS) {
            const float* p0 = X + ((size_t)(Ab0 * 4 + (kk >> 6)) * rpb + Ar0) * 64 + (kk & 63);
            const float* p1 = X + ((size_t)(Ab1 * 4 + (kk >> 6)) * rpb + Ar1) * 64 + (kk & 63);
            a0[0] = p0[0]; a0[1] = p0[1];
            a1[0] = p1[0]; a1[1] = p1[1];
        } else {
            const float* p0 = X + (size_t)rA0 * 256 + kk;
            const float* p1 = X + (size_t)rA1 * 256 + kk;
            a0[0] = p0[0]; a0[1] = p0[1];
            a1[0] = p1[0]; a1[1] = p1[1];
        }
        const float* pb0 = W + (size_t)(n0 + colc) * 256 + kk;
        const float* pb1 = W + (size_t)(n0 + 16 + colc) * 256 + kk;
        v2f b0; b0[0] = pb0[0]; b0[1] = pb0[1];
        v2f b1; b1[0] = pb1[0]; b1[1] = pb1[1];
        acc00 = WMMA_F32(a0, b0, acc00);
        acc01 = WMMA_F32(a0, b1, acc01);
        acc10 = WMMA_F32(a1, b0, acc10);
        acc11 = WMMA_F32(a1, b1, acc11);
    }

    const int rbase = 8 * (lane >> 4);
    for (int rt = 0; rt < 2; ++rt) {
        for (int ct = 0; ct < 2; ++ct) {
            const v8f acc = rt ? (ct ? acc11 : acc10) : (ct ? acc01 : acc00);
            const int n = n0 + ct * 16 + colc;
            const float bv = bias[n];
            for (int v = 0; v < 8; ++v) {
                const int rr = row0 + rt * 16 + rbase + v;
                if (rr >= nrows) continue;
                const float val = acc[v] + bv;
                if (OUT_HEADS) {
                    const int b = rr / rpb, r2 = rr % rpb;
                    Y[((size_t)(b * 4 + (n >> 6)) * rpb + r2) * 64 + (n & 63)] = val;
                } else {
                    Y[(size_t)rr * 256 + n] = val;
                }
            }
        }
    }
}

// ---------------------------------------------------------------------------
// Temporal-branch static-key logits (WMMA): per (b,h) this is a true GEMM
// [4096,64] x [64,32].  Writes scale*Qt.Ks^T + mask straight into tw cols 8..39
// as raw logits (normalized later by temporal_attn).
// Block: 256 threads = 8 waves; each wave: 16 rows x 32 static cols.
// ---------------------------------------------------------------------------
__global__ void qtks_logits(const float* __restrict__ Qt, const float* __restrict__ Ks,
                            const int* __restrict__ ms, float* __restrict__ tw)
{
    const int bh   = blockIdx.x >> 5;       // 0..63
    const int blk  = blockIdx.x & 31;       // 0..31 (128 rows each)
    const int wave = threadIdx.x >> 5;
    const int lane = threadIdx.x & 31;
    const int b    = bh >> 2;

    const int rows0 = blk * 128 + wave * 16;
    const int r    = lane & 15;
    const int ksel = (lane >> 4) * 2;
    const int colc = lane & 15;

    const float* A  = Qt + ((size_t)bh * 4096 + rows0) * 64;  // 16 x 64
    const float* K0 = Ks + (size_t)bh * 32 * 64;              // 32 x 64

    v8f acc0 = {}, acc1 = {};
    for (int k = 0; k < 64; k += 4) {
        const int kk = k + ksel;
        v2f a;  a[0]  = A[r * 64 + kk];             a[1]  = A[r * 64 + kk + 1];
        v2f b0; b0[0] = K0[colc * 64 + kk];         b0[1] = K0[colc * 64 + kk + 1];
        v2f b1; b1[0] = K0[(16 + colc) * 64 + kk];  b1[1] = K0[(16 + colc) * 64 + kk + 1];
        acc0 = WMMA_F32(a, b0, acc0);
        acc1 = WMMA_F32(a, b1, acc1);
    }

    const int rbase = 8 * (lane >> 4);
    for (int ct = 0; ct < 2; ++ct) {
        const int s = ct * 16 + colc;
        const float madd = (ms[b * 32 + s] == 1) ? 0.f : neg_inf();
        for (int v = 0; v < 8; ++v) {
            const int row = rows0 + rbase + v;      // = t*8 + m
            const float val = (ct ? acc1[v] : acc0[v]) * 0.125f + madd;
            tw[((size_t)bh * 4096 + row) * 40 + 8 + s] = val;
        }
    }
}

// ---------------------------------------------------------------------------
// Temporal branch: one wave per (b,h,t).
// Computes the 8x8 block-diagonal Qt.Kt^T logits on VALU, merges the raw
// static-key logits already in tw (from qtks_logits), softmaxes over 40,
// rewrites tw normalized, then P.V -> QKV_t (in place over Qt).
// ---------------------------------------------------------------------------
__global__ void temporal_attn(const float* __restrict__ Qt, const float* __restrict__ Kt,
                              const float* __restrict__ Vt, const float* __restrict__ Vs,
                              const int* __restrict__ mt,
                              float* __restrict__ tw, float* __restrict__ qkv_t)
{
    __shared__ float sQ[8 * 64];
    __shared__ float sK[8 * 64];
    __shared__ float sV[40 * 64];
    __shared__ float sL[8 * 40];

    const int idx = blockIdx.x;              // bh*512 + t
    const int t  = idx % 512;
    const int bh = idx / 512;
    const int b  = bh >> 2;
    const int lane = threadIdx.x;

    const size_t baseT = ((size_t)bh * 4096 + (size_t)t * 8) * 64;  // 8x64 slab
    const size_t baseS = (size_t)bh * 32 * 64;                      // 32x64 slab
    const size_t twrow = ((size_t)bh * 512 + t) * 8;                // tw row base

    for (int i = lane; i < 8 * 64; i += 32) {
        sQ[i] = Qt[baseT + i];
        sK[i] = Kt[baseT + i];
        sV[i] = Vt[baseT + i];
    }
    for (int i = lane; i < 32 * 64; i += 32)
        sV[8 * 64 + i] = Vs[baseS + i];
    for (int i = lane; i < 256; i += 32) {          // raw static logits from tw
        const int row = i >> 5, col = i & 31;
        sL[row * 40 + 8 + col] = tw[(twrow + row) * 40 + 8 + col];
    }
    __syncthreads();

    const float scale = 0.125f;  // 1/sqrt(64)
    for (int e = lane; e < 64; e += 32) {           // 8x8 temporal logits
        const int row = e >> 3, col = e & 7;
        float s = 0.f;
        for (int k = 0; k < 64; ++k) s += sQ[row * 64 + k] * sK[col * 64 + k];
        const int mval = mt[((size_t)b * 512 + t) * 8 + col];
        sL[row * 40 + col] = s * scale + ((mval == 1) ? 0.f : neg_inf());
    }
    __syncthreads();

    if (lane < 8) {
        const int row = lane;
        float mx = neg_inf();
        for (int c = 0; c < 40; ++c) mx = fmaxf(mx, sL[row * 40 + c]);
        float sum = 0.f;
        for (int c = 0; c < 40; ++c) {
            const float e = expf(sL[row * 40 + c] - mx);
            sL[row * 40 + c] = e;
            sum += e;
        }
        const float inv = 1.f / sum;
        for (int c = 0; c < 40; ++c) {
            const float p = sL[row * 40 + c] * inv;
            sL[row * 40 + c] = p;
            tw[(twrow + row) * 40 + c] = p;
        }
    }
    __syncthreads();

    for (int e = lane; e < 512; e += 32) {          // P.V : 8x64
        const int row = e >> 6, col = e & 63;
        float s = 0.f;
        for (int j = 0; j < 40; ++j) s += sL[row * 40 + j] * sV[j * 64 + col];
        qkv_t[baseT + e] = s;
    }
}

// ---------------------------------------------------------------------------
// Static branch pass 1 (WMMA): raw logits sw[b,h,32,4128] = scale*Qs.K^T + mask.
// 8 waves per block, each wave: 1 col-tile (16 keys) x 32 query rows.
// ---------------------------------------------------------------------------
__global__ void static_logits(const float* __restrict__ Qs, const float* __restrict__ Kt,
                              const float* __restrict__ Ks, const int* __restrict__ mt,
                              const int* __restrict__ ms, float* __restrict__ sw)
{
    const int CHUNKS = 33;                     // ceil(258 col tiles / 8 waves)
    const int chunk = blockIdx.x % CHUNKS;
    const int bh = blockIdx.x / CHUNKS;
    const int h = bh & 3, b = bh >> 2;
    const int wave = threadIdx.x >> 5;
    const int lane = threadIdx.x & 31;
    const int ct = chunk * 8 + wave;
    if (ct >= 258) return;                     // whole wave idles; WMMA waves stay full-EXEC

    const int c0 = ct * 16;
    const int colc = lane & 15;
    const int ksel = (lane >> 4) * 2;
    const int c = c0 + colc;
    const int cc = (c > 4127) ? 4127 : c;

    const float* Krow = (cc < 4096)
        ? Kt + (((size_t)(b * 4 + h) * 4096) + cc) * 64
        : Ks + (((size_t)(b * 4 + h) * 32) + (cc - 4096)) * 64;
    const float* Q0 = Qs + (size_t)(b * 4 + h) * 32 * 64;
    const int r = lane & 15;

    v8f acc0 = {};
    v8f acc1 = {};
    for (int k = 0; k < 64; k += 4) {
        const int kk = k + ksel;
        v2f bv; bv[0] = Krow[kk];               bv[1] = Krow[kk + 1];
        v2f a0; a0[0] = Q0[r * 64 + kk];        a0[1] = Q0[r * 64 + kk + 1];
        v2f a1; a1[0] = Q0[(r + 16) * 64 + kk]; a1[1] = Q0[(r + 16) * 64 + kk + 1];
        acc0 = WMMA_F32(a0, bv, acc0);
        acc1 = WMMA_F32(a1, bv, acc1);
    }

    if (c < 4128) {
        const int mval = (c < 4096) ? mt[(size_t)b * 4096 + c] : ms[b * 32 + (c - 4096)];
        const float madd = (mval == 1) ? 0.f : neg_inf();
        const size_t base = ((size_t)(b * 4 + h) * 32) * 4128 + c;
        const int rb = 8 * (lane >> 4);
        for (int v = 0; v < 8; ++v) {
            sw[base + (size_t)(rb + v) * 4128]      = acc0[v] * 0.125f + madd;
            sw[base + (size_t)(rb + v + 16) * 4128] = acc1[v] * 0.125f + madd;
        }
    }
}

// ---------------------------------------------------------------------------
// Static branch pass 2: per (b,h,q) block of 256 threads.
// Softmax over 4128 (normalizes sw in place -> required output), then P.V.
// ---------------------------------------------------------------------------
__global__ void static_softmax_pv(float* __restrict__ sw, const float* __restrict__ Vt,
                                  const float* __restrict__ Vs, float* __restrict__ qkvs)
{
    const int q = blockIdx.x & 31;
    const int bh = blockIdx.x >> 5;
    const int h = bh & 3, b = bh >> 2;
    float* row = sw + (((size_t)(b * 4 + h) * 32) + q) * 4128;

    __shared__ float red[256];
    const int tid = threadIdx.x;

    float mx = neg_inf();
    for (int i = tid; i < 4128; i += 256) mx = fmaxf(mx, row[i]);
    red[tid] = mx;
    __syncthreads();
    for (int s = 128; s > 0; s >>= 1) {
        if (tid < s) red[tid] = fmaxf(red[tid], red[tid + s]);
        __syncthreads();
    }
    mx = red[0];
    __syncthreads();

    float sum = 0.f;
    for (int i = tid; i < 4128; i += 256) {
        const float e = expf(row[i] - mx);
        row[i] = e;
        sum += e;
    }
    red[tid] = sum;
    __syncthreads();
    for (int s = 128; s > 0; s >>= 1) {
        if (tid < s) red[tid] += red[tid + s];
        __syncthreads();
    }
    const float inv = 1.f / red[0];

    for (int i = tid; i < 4128; i += 256) row[i] *= inv;
    __syncthreads();

    const int hd = tid & 63, stripe = tid >> 6;
    float acc = 0.f;
    const size_t vtb = ((size_t)(b * 4 + h) * 4096) * 64;
    const size_t vsb = ((size_t)(b * 4 + h) * 32) * 64;
    for (int j = stripe; j < 4096; j += 4)
        acc += row[j] * Vt[vtb + (size_t)j * 64 + hd];
    for (int j = 4096 + stripe; j < 4128; j += 4)
        acc += row[j] * Vs[vsb + (size_t)(j - 4096) * 64 + hd];
    red[tid] = acc;
    __syncthreads();
    if (stripe == 0)
        qkvs[(((size_t)(b * 4 + h) * 32) + q) * 64 + hd] =
            red[hd] + red[64 + hd] + red[128 + hd] + red[192 + hd];
}

// ---------------------------------------------------------------------------
extern "C" void kernel_launch(void* const* d_in, const int* in_sizes, int n_in,
                              void* d_out, int out_size, void* d_ws, size_t ws_size,
                              hipStream_t stream)
{
    (void)in_sizes; (void)n_in; (void)out_size; (void)ws_size;

    const float* q_temporal = (const float*)d_in[0];
    const float* q_static   = (const float*)d_in[1];
    const float* k_temporal = (const float*)d_in[2];
    const float* v_temporal = (const float*)d_in[3];
    const float* k_static   = (const float*)d_in[4];
    const float* v_static   = (const float*)d_in[5];
    const float* q_w = (const float*)d_in[6];
    const float* q_b = (const float*)d_in[7];
    const float* k_w = (const float*)d_in[8];
    const float* k_b = (const float*)d_in[9];
    const float* v_w = (const float*)d_in[10];
    const float* v_b = (const float*)d_in[11];
    const float* o_w = (const float*)d_in[12];
    const float* o_b = (const float*)d_in[13];
    const int*   mt  = (const int*)d_in[14];
    // d_in[15] = mask_fcst (unused by reference)
    const int*   ms  = (const int*)d_in[16];

    // d_out partition (floats): out_t | out_s | tw | sw
    float* out   = (float*)d_out;
    float* out_t = out;                         // 16,777,216
    float* out_s = out + 16777216;              //    131,072
    float* tw    = out + 16908288;              // 10,485,760
    float* sw    = out + 27394048;              //  8,454,144

    // workspace partition (floats)
    float* wsf = (float*)d_ws;
    float* Qt   = wsf;                          // 16,777,216 (reused for QKV_t)
    float* Kt   = Qt + 16777216;
    float* Vt   = Kt + 16777216;
    float* Qs   = Vt + 16777216;                // 131,072 each below
    float* Ks   = Qs + 131072;
    float* Vs   = Ks + 131072;
    float* QKVs = Vs + 131072;

    dim3 blk(256);

    // 1) projections (fp32 WMMA), outputs in head-split layout
    gemm256<0, 1><<<2048, blk, 0, stream>>>(q_temporal, q_w, q_b, Qt, 65536, 4096);
    gemm256<0, 1><<<2048, blk, 0, stream>>>(k_temporal, k_w, k_b, Kt, 65536, 4096);
    gemm256<0, 1><<<2048, blk, 0, stream>>>(v_temporal, v_w, v_b, Vt, 65536, 4096);
    gemm256<0, 1><<<16,   blk, 0, stream>>>(q_static,   q_w, q_b, Qs, 512,   32);
    gemm256<0, 1><<<16,   blk, 0, stream>>>(k_static,   k_w, k_b, Ks, 512,   32);
    gemm256<0, 1><<<16,   blk, 0, stream>>>(v_static,   v_w, v_b, Vs, 512,   32);

    // 2) temporal branch: WMMA Qt.Ks^T raw logits into tw, then per-(b,h,t)
    //    block-diagonal logits + softmax + P.V (QKV_t overwrites Qt in place;
    //    qtks_logits reads Qt first — same-stream order guarantees safety)
    qtks_logits<<<64 * 32, blk, 0, stream>>>(Qt, Ks, ms, tw);
    temporal_attn<<<32768, 32, 0, stream>>>(Qt, Kt, Vt, Vs, mt, tw, Qt);

    // 3) static branch: WMMA logits -> sw, then in-place softmax + P.V
    static_logits<<<64 * 33, blk, 0, stream>>>(Qs, Kt, Ks, mt, ms, sw);
    static_softmax_pv<<<2048, blk, 0, stream>>>(sw, Vt, Vs, QKVs);

    // 4) output projections
    gemm256<1, 0><<<2048, blk, 0, stream>>>(Qt,   o_w, o_b, out_t, 65536, 4096);
    gemm256<1, 0><<<16,   blk, 0, stream>>>(QKVs, o_w, o_b, out_s, 512,   32);
}